// CountScaledLMHeadLoss_55353538511245
// MI455X (gfx1250) — compile-verified
//
#include <hip/hip_runtime.h>
#include <hip/hip_bf16.h>
#include <stdint.h>

// ---------------- problem constants (from reference) ----------------
#define B_DIM 32
#define L_DIM 65536
#define T_DIM 32
#define F_DIM 4
// TEMP = 1.0, DELTA = 1.0, LAMBDA = 1.0

// ---------------- launch geometry ----------------
#define TPB      256          // 8 waves of 32 on CDNA5
#define WAVES    (TPB / 32)
#define NBLK_A   128          // blocks per example, logits kernel
#define NBLK_B   64           // blocks per example, gt_tracks kernel
// per-wave work in kernel B: TL / (NBLK_B*WAVES) = 2097152/512 = 4096 floats
#define B_ITERS  32           // 32 iters * 128 floats/iter = 4096

typedef __attribute__((ext_vector_type(4))) float vf4;
typedef __attribute__((ext_vector_type(2))) float v2f;
typedef __attribute__((ext_vector_type(8))) float v8f;

#if defined(__HIP_DEVICE_COMPILE__)
#if defined(__has_builtin)
#if __has_builtin(__builtin_amdgcn_wmma_f32_16x16x4_f32)
#define USE_WMMA_REDUCE 1
#else
#warning "wmma_f32_16x16x4_f32 builtin not available; falling back to VALU reduction"
#endif
#endif
#endif

// =====================================================================
// Kernel A: per-position softmax / KL / entropy / huber statistics.
// grid = (NBLK_A, B), block = TPB.  Writes 6 partials per (b, block).
// =====================================================================
__global__ __launch_bounds__(TPB) void csl_logits_partial(
    const float* __restrict__ ref_onehot,     // (B,L,4)
    const uint8_t* __restrict__ mask,         // (B,L) bool
    const float* __restrict__ tlog,           // (B,L,4)
    const float* __restrict__ slog,           // (B,L,4)
    float* __restrict__ wsA)                  // (B, NBLK_A, 6)
{
    const int b   = blockIdx.y;
    const int tid = threadIdx.x;
    const long base = (long)b * L_DIM;

    float acc[6] = {0.f, 0.f, 0.f, 0.f, 0.f, 0.f};
    // acc: kl_num, mask_cnt, ent_num, align_cnt, ref_num, ref_cnt

    const vf4* t4 = (const vf4*)tlog;
    const vf4* s4 = (const vf4*)slog;
    const vf4* r4 = (const vf4*)ref_onehot;

    for (int l = blockIdx.x * TPB + tid; l < L_DIM; l += TPB * NBLK_A) {
        const long idx = base + l;
        vf4 tv = __builtin_nontemporal_load(t4 + idx);
        vf4 sv = __builtin_nontemporal_load(s4 + idx);
        vf4 rv = __builtin_nontemporal_load(r4 + idx);
        uint8_t mk = __builtin_nontemporal_load(mask + idx);

        float ta[4] = {tv.x, tv.y, tv.z, tv.w};
        float sa[4] = {sv.x, sv.y, sv.z, sv.w};
        float ra[4] = {rv.x, rv.y, rv.z, rv.w};

        // log_softmax (TEMP == 1)
        float mt = fmaxf(fmaxf(ta[0], ta[1]), fmaxf(ta[2], ta[3]));
        float ms = fmaxf(fmaxf(sa[0], sa[1]), fmaxf(sa[2], sa[3]));
        float zt = 0.f, zs = 0.f;
#pragma unroll
        for (int j = 0; j < 4; ++j) { zt += expf(ta[j] - mt); zs += expf(sa[j] - ms); }
        float lse_t = mt + logf(zt);
        float lse_s = ms + logf(zs);

        float kl = 0.f, Hp = 0.f, Hq = 0.f;
        float lp[4], lq[4];
#pragma unroll
        for (int j = 0; j < 4; ++j) {
            lp[j] = ta[j] - lse_t;
            lq[j] = sa[j] - lse_s;
            float p = expf(lp[j]);
            float q = expf(lq[j]);
            kl += p * (lp[j] - lq[j]);
            Hp -= p * lp[j];
            Hq -= q * lq[j];
        }

        // argmax with first-occurrence tie-break (matches jnp.argmax)
        int at = 0, as = 0, ri = 0;
        float bt = ta[0], bs = sa[0], br = ra[0];
#pragma unroll
        for (int j = 1; j < 4; ++j) {
            if (ta[j] > bt) { bt = ta[j]; at = j; }
            if (sa[j] > bs) { bs = sa[j]; as = j; }
            if (ra[j] > br) { br = ra[j]; ri = j; }
        }

        float mf = mk ? 1.f : 0.f;
        acc[0] += kl * mf;
        acc[1] += mf;

        float am = (at == as) ? mf : 0.f;
        float ed = Hq - Hp;
        acc[2] += ed * ed * am;
        acc[3] += am;

        float rm = (at == ri) ? mf : 0.f;
        float gap = lq[ri] - lp[ri];          // (-log_p_ref) - (-log_q_ref)
        float pos = fmaxf(gap, 0.f);
        float hub = (pos <= 1.f) ? (0.5f * pos * pos) : (pos - 0.5f);  // DELTA=1
        acc[4] += hub * rm;
        acc[5] += rm;
    }

    // wave reduce (wave32), then cross-wave via LDS — fixed order, deterministic
#pragma unroll
    for (int k = 0; k < 6; ++k)
        for (int off = 16; off > 0; off >>= 1)
            acc[k] += __shfl_xor(acc[k], off, 32);

    __shared__ float sm[WAVES][6];
    const int wid = tid >> 5, lane = tid & 31;
    if (lane == 0)
#pragma unroll
        for (int k = 0; k < 6; ++k) sm[wid][k] = acc[k];
    __syncthreads();
    if (tid == 0) {
        float* dst = wsA + ((long)b * NBLK_A + blockIdx.x) * 6;
#pragma unroll
        for (int k = 0; k < 6; ++k) {
            float s = 0.f;
            for (int w = 0; w < WAVES; ++w) s += sm[w][k];
            dst[k] = s;
        }
    }
}

// =====================================================================
// Kernel B: per-example sum of clip(gt_tracks, 0).  268 MB stream.
// Accumulation routed through the matrix pipe: D = ones(16x4) x B + C
// gives column-sums of 64 f32 per V_WMMA_F32_16X16X4_F32 issue; since we
// sum all 16 column-sums at the end, the packing of loaded values into
// the B operand's two VGPRs is irrelevant.  Uniform control flow keeps
// EXEC all-ones as WMMA requires.
// grid = (NBLK_B, B), block = TPB.
// =====================================================================
__global__ __launch_bounds__(TPB) void csl_tracks_partial(
    const float* __restrict__ gt,             // (B,T,L) contiguous per example
    float* __restrict__ wsB)                  // (B, NBLK_B)
{
    const int b    = blockIdx.y;
    const int tid  = threadIdx.x;
    const int lane = tid & 31;
    const int wid  = tid >> 5;
    const long TL  = (long)T_DIM * L_DIM;                  // 2,097,152
    const int gwave = blockIdx.x * WAVES + wid;            // 0..511
    const float* p = gt + (long)b * TL + (long)gwave * (B_ITERS * 128) + lane * 4;

    float wave_sum;
#if defined(USE_WMMA_REDUCE)
    v2f a_ones = {1.f, 1.f};
    v8f c0 = {}; v8f c1 = {};
#pragma unroll 4
    for (int it = 0; it < B_ITERS; ++it) {
        vf4 d = __builtin_nontemporal_load((const vf4*)(p + it * 128));
        v2f b0 = { fmaxf(d.x, 0.f), fmaxf(d.y, 0.f) };
        v2f b1 = { fmaxf(d.z, 0.f), fmaxf(d.w, 0.f) };
        c0 = __builtin_amdgcn_wmma_f32_16x16x4_f32(false, a_ones, false, b0,
                                                   (short)0, c0, false, false);
        c1 = __builtin_amdgcn_wmma_f32_16x16x4_f32(false, a_ones, false, b1,
                                                   (short)0, c1, false, false);
    }
    // c[0] at lane n holds colsum_{n&15} (rows M=0 / M=8 duplicate the value),
    // so the 32-lane sum is exactly 2x the wave total -> *0.5 is exact.
    float v = c0[0] + c1[0];
    for (int off = 16; off > 0; off >>= 1) v += __shfl_xor(v, off, 32);
    wave_sum = v * 0.5f;
#else
    float acc = 0.f;
#pragma unroll 4
    for (int it = 0; it < B_ITERS; ++it) {
        vf4 d = __builtin_nontemporal_load((const vf4*)(p + it * 128));
        acc += fmaxf(d.x, 0.f) + fmaxf(d.y, 0.f) + fmaxf(d.z, 0.f) + fmaxf(d.w, 0.f);
    }
    for (int off = 16; off > 0; off >>= 1) acc += __shfl_xor(acc, off, 32);
    wave_sum = acc;
#endif

    __shared__ float sm[WAVES];
    if (lane == 0) sm[wid] = wave_sum;
    __syncthreads();
    if (tid == 0) {
        float s = 0.f;
        for (int w = 0; w < WAVES; ++w) s += sm[w];
        wsB[(long)b * NBLK_B + blockIdx.x] = s;
    }
}

// =====================================================================
// Kernel C: finalize.  One wave: thread b owns example b (fixed-order
// serial sums -> deterministic), then 32-lane reduce -> scalar.
// =====================================================================
__global__ __launch_bounds__(32) void csl_finalize(
    const float* __restrict__ wsA,
    const float* __restrict__ wsB,
    float* __restrict__ out)
{
    const int b = threadIdx.x;                 // 32 threads == B_DIM
    float s[6] = {0.f, 0.f, 0.f, 0.f, 0.f, 0.f};
    const float* pa = wsA + (long)b * NBLK_A * 6;
    for (int k = 0; k < NBLK_A; ++k)
#pragma unroll
        for (int j = 0; j < 6; ++j) s[j] += pa[k * 6 + j];

    float N = 0.f;
    const float* pb = wsB + (long)b * NBLK_B;
    for (int k = 0; k < NBLK_B; ++k) N += pb[k];

    float kl_pe  = s[0] / fmaxf(s[1], 1.f);
    float ent_pe = s[2] / fmaxf(s[3], 1.f);
    float ref_pe = s[4] / fmaxf(s[5], 1.f);
    float coeff  = log1pf(fmaxf(N, 0.f));
    float contrib = coeff * (kl_pe + ent_pe + ref_pe);

    for (int off = 16; off > 0; off >>= 1) contrib += __shfl_xor(contrib, off, 32);
    if (b == 0) out[0] = 1.0f /*LAMBDA*/ * (contrib / (float)B_DIM);
}

// =====================================================================
// Launcher.  Inputs (setup_inputs order):
//   0 ref_onehot (B,L,4) f32 | 1 mask (B,L) bool(u8) | 2 teacher (B,L,4) f32
//   3 student (B,L,4) f32    | 4 gt_tracks (B,T,L) f32
// d_ws layout: [B*NBLK_A*6 floats wsA][B*NBLK_B floats wsB]  (~107 KB)
// Every ws slot read by kernel C is written earlier in the same call; no
// atomics, no persistent state -> deterministic under graph replay.
// =====================================================================
extern "C" void kernel_launch(void* const* d_in, const int* in_sizes, int n_in,
                              void* d_out, int out_size, void* d_ws, size_t ws_size,
                              hipStream_t stream) {
    (void)in_sizes; (void)n_in; (void)out_size; (void)ws_size;
    const float*   ref_onehot = (const float*)d_in[0];
    const uint8_t* mask       = (const uint8_t*)d_in[1];
    const float*   tlog       = (const float*)d_in[2];
    const float*   slog       = (const float*)d_in[3];
    const float*   gt         = (const float*)d_in[4];

    float* wsA = (float*)d_ws;
    float* wsB = wsA + (long)B_DIM * NBLK_A * 6;

    csl_logits_partial<<<dim3(NBLK_A, B_DIM), TPB, 0, stream>>>(
        ref_onehot, mask, tlog, slog, wsA);
    csl_tracks_partial<<<dim3(NBLK_B, B_DIM), TPB, 0, stream>>>(gt, wsB);
    csl_finalize<<<1, 32, 0, stream>>>(wsA, wsB, (float*)d_out);
}